// TrustGCN_18330920419683
// MI455X (gfx1250) — compile-verified
//
#include <hip/hip_runtime.h>
#include <hip/hip_bf16.h>
#include <math.h>

// ---------------------------------------------------------------------------
// GCN forward for MI455X (gfx1250).
// Heavy part: edge gather/scatter (L2-resident node features, HBM-streamed
// edge indices) using hardware global_atomic_add_f32. Dense per-layer
// transforms use V_WMMA_F32_16X16X4_F32 (exact f32 WMMA) with branch-free
// fragment loads and LDS-staged coalesced stores.
// ---------------------------------------------------------------------------

typedef __attribute__((ext_vector_type(2))) float v2f;
typedef __attribute__((ext_vector_type(8))) float v8f;

#if defined(__has_builtin)
#if __has_builtin(__builtin_amdgcn_wmma_f32_16x16x4_f32)
#define HAVE_WMMA_F32 1
#endif
#endif
#ifndef HAVE_WMMA_F32
#define HAVE_WMMA_F32 0
#endif

__device__ __forceinline__ void atomic_add_f32(float* p, float v) {
#if defined(__gfx1250__)
  // Guarantee the hardware FP32 atomic (no CAS loop): no-return form,
  // tracked with STOREcnt.
  asm volatile("global_atomic_add_f32 %0, %1, off" : : "v"(p), "v"(v) : "memory");
#else
  __hip_atomic_fetch_add(p, v, __ATOMIC_RELAXED, __HIP_MEMORY_SCOPE_AGENT);
#endif
}

__device__ __forceinline__ float elu1(float x) {
  return x > 0.0f ? x : (expf(x) - 1.0f);
}

// ---------------------------------------------------------------------------
// 1) degree counting (+ optional int64 -> int32 index conversion)
// ---------------------------------------------------------------------------
__global__ void k_zero_i32(int* __restrict__ p, int n) {
  int i = blockIdx.x * blockDim.x + threadIdx.x;
  if (i < n) p[i] = 0;
}

template <bool CONVERT>
__global__ void k_degree(const long long* __restrict__ src,
                         const long long* __restrict__ dst,
                         int* __restrict__ deg,
                         int* __restrict__ s32, int* __restrict__ d32, int E) {
  int e = blockIdx.x * blockDim.x + threadIdx.x;
  if (e >= E) return;
  int d = (int)__builtin_nontemporal_load(&dst[e]);
  atomicAdd(deg + d, 1);
  if (CONVERT) {
    int s = (int)__builtin_nontemporal_load(&src[e]);
    __builtin_nontemporal_store(s, &s32[e]);
    __builtin_nontemporal_store(d, &d32[e]);
  }
}

// deg (int, edges into node) -> dinv = rsqrt(deg + 1 self-loop), in place
__global__ void k_dinv(int* __restrict__ degbuf, int n) {
  int i = blockIdx.x * blockDim.x + threadIdx.x;
  if (i >= n) return;
  int c = degbuf[i];
  float* out = reinterpret_cast<float*>(degbuf);
  out[i] = rsqrtf((float)c + 1.0f);
}

// ---------------------------------------------------------------------------
// 2) dense transform  out[N,F_OUT] = (ELU?)(in[N,F_IN]) @ W[F_IN,F_OUT]
//    One wave32 per 16-row tile, V_WMMA_F32_16X16X4_F32, K padded to mult of 4,
//    W zero-padded to 16 columns. Fragment layouts (ISA 7.12.2):
//      A 16x4 : lane = 16*(k>=2) + m, reg = k&1
//      B 4x16 : lane = 16*(k>=2) + n, reg = k&1
//      C/D    : row = 8*(lane>=16) + reg, col = lane&15
//    Loads are unconditional (clamped index) + v_cndmask masking; D fragment
//    is staged through LDS and written back as coalesced float4 stores.
// ---------------------------------------------------------------------------
#if HAVE_WMMA_F32
template <int F_IN, int F_OUT, bool ELU_IN>
__global__ void k_transform(const float* __restrict__ in,
                            const float* __restrict__ W,
                            float* __restrict__ out, int n_nodes) {
  __shared__ float tile[8][256];  // one 16x16 f32 tile per wave (8 waves/block)
  const int wslot = threadIdx.x >> 5;
  const int wave = blockIdx.x * (blockDim.x >> 5) + wslot;
  const int lane = threadIdx.x & 31;
  const int m = lane & 15;      // A row within tile / B,D column
  const int khalf = lane >> 4;  // 0: k in {0,1}; 1: k in {2,3}
  const int node = wave * 16 + m;
  const bool node_ok = node < n_nodes;
  const int nodec = node_ok ? node : 0;
  const bool m_ok = (m < F_OUT);
  const int mc = m_ok ? m : 0;

  v8f c = {0.f, 0.f, 0.f, 0.f, 0.f, 0.f, 0.f, 0.f};
  constexpr int KSTEPS = (F_IN + 3) / 4;
#pragma unroll
  for (int ks = 0; ks < KSTEPS; ++ks) {
    v2f a, b;
#pragma unroll
    for (int r = 0; r < 2; ++r) {
      const int k = ks * 4 + khalf * 2 + r;
      const bool k_ok = (k < F_IN);
      const int kc = k_ok ? k : 0;
      float av = in[(size_t)nodec * F_IN + kc];   // unconditional load
      if (ELU_IN) av = elu1(av);
      av = (k_ok && node_ok) ? av : 0.0f;         // mask via cndmask
      float bv = W[kc * F_OUT + mc];              // unconditional load
      bv = (k_ok && m_ok) ? bv : 0.0f;
      a[r] = av;
      b[r] = bv;
    }
    c = __builtin_amdgcn_wmma_f32_16x16x4_f32(false, a, false, b, (short)0, c,
                                              false, false);
  }

  // stage D fragment to LDS (wave-private tile; DS ops in-order within wave)
  float* t = tile[wslot];
#pragma unroll
  for (int r = 0; r < 8; ++r) t[(khalf * 8 + r) * 16 + m] = c[r];

  const long long base = (long long)wave * 16;
  if constexpr ((F_OUT % 4) == 0) {
    constexpr int CPR = F_OUT / 4;        // float4 chunks per row
    constexpr int CHUNKS = 16 * CPR;      // 32 (F=8) or 64 (F=16)
#pragma unroll
    for (int it = 0; it < CHUNKS / 32; ++it) {
      const int chunk = it * 32 + lane;
      const int row = chunk / CPR;
      const int c4 = chunk % CPR;
      if (base + row < n_nodes) {
        float4 v = *reinterpret_cast<const float4*>(&t[row * 16 + c4 * 4]);
        *reinterpret_cast<float4*>(&out[(base + row) * F_OUT + c4 * 4]) = v;
      }
    }
  } else {  // F_OUT == 2: 32 floats, one per lane
    const int row = lane >> 1;
    const int col = lane & 1;
    if (base + row < n_nodes)
      out[(base + row) * F_OUT + col] = t[row * 16 + col];
  }
}
#else
// Scalar fallback (same launch geometry: threads >= 2*N guaranteed)
template <int F_IN, int F_OUT, bool ELU_IN>
__global__ void k_transform(const float* __restrict__ in,
                            const float* __restrict__ W,
                            float* __restrict__ out, int n_nodes) {
  int i = blockIdx.x * blockDim.x + threadIdx.x;
  if (i >= n_nodes) return;
  float a[F_IN];
#pragma unroll
  for (int k = 0; k < F_IN; ++k) {
    float v = in[(long long)i * F_IN + k];
    a[k] = ELU_IN ? elu1(v) : v;
  }
#pragma unroll
  for (int n = 0; n < F_OUT; ++n) {
    float acc = 0.f;
#pragma unroll
    for (int k = 0; k < F_IN; ++k) acc = fmaf(a[k], W[k * F_OUT + n], acc);
    out[(long long)i * F_OUT + n] = acc;
  }
}
#endif

// ---------------------------------------------------------------------------
// 3) self-loop + bias init:  out[n] = b + dinv[n]^2 * h[n]
// ---------------------------------------------------------------------------
template <int F>
__global__ void k_self_init(const float* __restrict__ hin,
                            const float* __restrict__ bias,
                            const float* __restrict__ dinv,
                            float* __restrict__ out, int n) {
  int i = blockIdx.x * blockDim.x + threadIdx.x;
  if (i >= n) return;
  float di = dinv[i];
  float w = di * di;
#pragma unroll
  for (int f = 0; f < F; ++f)
    out[(long long)i * F + f] = bias[f] + w * hin[(long long)i * F + f];
}

// ---------------------------------------------------------------------------
// 4) edge scatter:  out[dst] += dinv[src]*dinv[dst] * h[src]
// ---------------------------------------------------------------------------
template <int F, typename IdxT>
__global__ void k_edge_agg(const IdxT* __restrict__ src,
                           const IdxT* __restrict__ dst,
                           const float* __restrict__ dinv,
                           const float* __restrict__ hin,
                           float* __restrict__ out, int E) {
  int e = blockIdx.x * blockDim.x + threadIdx.x;
  if (e >= E) return;
  int s = (int)__builtin_nontemporal_load(&src[e]);
  int d = (int)__builtin_nontemporal_load(&dst[e]);
  float w = dinv[s] * dinv[d];
  const float* hp = hin + (long long)s * F;
  float* op = out + (long long)d * F;
  if constexpr ((F % 4) == 0) {
    const float4* h4 = reinterpret_cast<const float4*>(hp);
#pragma unroll
    for (int q = 0; q < F / 4; ++q) {
      float4 v = h4[q];
      atomic_add_f32(op + 4 * q + 0, w * v.x);
      atomic_add_f32(op + 4 * q + 1, w * v.y);
      atomic_add_f32(op + 4 * q + 2, w * v.z);
      atomic_add_f32(op + 4 * q + 3, w * v.w);
    }
  } else {
#pragma unroll
    for (int f = 0; f < F; ++f) atomic_add_f32(op + f, w * hp[f]);
  }
}

// ---------------------------------------------------------------------------
// 5) log-softmax over 2 classes, in place
// ---------------------------------------------------------------------------
__global__ void k_logsoftmax2(float* __restrict__ out, int n) {
  int i = blockIdx.x * blockDim.x + threadIdx.x;
  if (i >= n) return;
  float a = out[2 * i], b = out[2 * i + 1];
  float m = fmaxf(a, b);
  float lse = m + logf(expf(a - m) + expf(b - m));
  out[2 * i] = a - lse;
  out[2 * i + 1] = b - lse;
}

// ---------------------------------------------------------------------------
// host side
// ---------------------------------------------------------------------------
static inline int cdiv_i(long long a, long long b) { return (int)((a + b - 1) / b); }

extern "C" void kernel_launch(void* const* d_in, const int* in_sizes, int n_in,
                              void* d_out, int out_size, void* d_ws, size_t ws_size,
                              hipStream_t stream) {
  const float* x = (const float*)d_in[0];
  const long long* ei = (const long long*)d_in[1];
  // d_in[2] (laplacian_index) / d_in[3] (laplacian_weight) unused by reference
  const float* W1 = (const float*)d_in[4];
  const float* b1 = (const float*)d_in[5];
  const float* W2 = (const float*)d_in[6];
  const float* b2 = (const float*)d_in[7];
  const float* W3 = (const float*)d_in[8];
  const float* b3 = (const float*)d_in[9];
  const float* W4 = (const float*)d_in[10];
  const float* b4 = (const float*)d_in[11];

  const int N = in_sizes[0] / 5;
  const int E = in_sizes[1] / 2;
  const long long* src64 = ei;
  const long long* dst64 = ei + E;

  // workspace carve-out (256B aligned slots)
  char* ws = (char*)d_ws;
  size_t off = 0;
  auto alloc = [&](size_t bytes) -> char* {
    char* p = ws + off;
    off = (off + bytes + 255) & ~(size_t)255;
    return p;
  };
  float* dinv = (float*)alloc((size_t)N * 4);     // deg(int) then rsqrt (float)
  float* hA = (float*)alloc((size_t)N * 16 * 4);  // ping
  float* hB = (float*)alloc((size_t)N * 16 * 4);  // pong
  int* s32 = nullptr;
  int* d32 = nullptr;
  if (off + 2 * (size_t)E * 4 <= ws_size) {
    s32 = (int*)alloc((size_t)E * 4);
    d32 = (int*)alloc((size_t)E * 4);
  }

  const int TB = 256;
  const int gN = cdiv_i(N, TB);
  const int gE = cdiv_i(E, TB);
  // one wave32 per 16-row tile; 8 waves per block
  const int tiles = cdiv_i(N, 16);
  const int gT = cdiv_i(tiles, 8);

  // --- degree / dinv ---
  k_zero_i32<<<gN, TB, 0, stream>>>((int*)dinv, N);
  if (s32)
    k_degree<true><<<gE, TB, 0, stream>>>(src64, dst64, (int*)dinv, s32, d32, E);
  else
    k_degree<false><<<gE, TB, 0, stream>>>(src64, dst64, (int*)dinv, nullptr, nullptr, E);
  k_dinv<<<gN, TB, 0, stream>>>((int*)dinv, N);

  // --- layer 1: 5 -> 8 ---
  k_transform<5, 8, false><<<gT, TB, 0, stream>>>(x, W1, hA, N);
  k_self_init<8><<<gN, TB, 0, stream>>>(hA, b1, dinv, hB, N);
  if (s32) k_edge_agg<8, int><<<gE, TB, 0, stream>>>(s32, d32, dinv, hA, hB, E);
  else     k_edge_agg<8, long long><<<gE, TB, 0, stream>>>(src64, dst64, dinv, hA, hB, E);

  // --- layer 2: 8 -> 16 (ELU on input) ---
  k_transform<8, 16, true><<<gT, TB, 0, stream>>>(hB, W2, hA, N);
  k_self_init<16><<<gN, TB, 0, stream>>>(hA, b2, dinv, hB, N);
  if (s32) k_edge_agg<16, int><<<gE, TB, 0, stream>>>(s32, d32, dinv, hA, hB, E);
  else     k_edge_agg<16, long long><<<gE, TB, 0, stream>>>(src64, dst64, dinv, hA, hB, E);

  // --- layer 3: 16 -> 8 ---
  k_transform<16, 8, true><<<gT, TB, 0, stream>>>(hB, W3, hA, N);
  k_self_init<8><<<gN, TB, 0, stream>>>(hA, b3, dinv, hB, N);
  if (s32) k_edge_agg<8, int><<<gE, TB, 0, stream>>>(s32, d32, dinv, hA, hB, E);
  else     k_edge_agg<8, long long><<<gE, TB, 0, stream>>>(src64, dst64, dinv, hA, hB, E);

  // --- layer 4: 8 -> 2, aggregate straight into d_out ---
  float* outp = (float*)d_out;
  k_transform<8, 2, true><<<gT, TB, 0, stream>>>(hB, W4, hA, N);
  k_self_init<2><<<gN, TB, 0, stream>>>(hA, b4, dinv, outp, N);
  if (s32) k_edge_agg<2, int><<<gE, TB, 0, stream>>>(s32, d32, dinv, hA, outp, E);
  else     k_edge_agg<2, long long><<<gE, TB, 0, stream>>>(src64, dst64, dinv, hA, outp, E);

  // --- log-softmax ---
  k_logsoftmax2<<<gN, TB, 0, stream>>>(outp, N);
}